// LIF_Activation_2319282339996
// MI455X (gfx1250) — compile-verified
//
#include <hip/hip_runtime.h>
#include <cstdint>

// LIF scan: out[b,t,c] = step(Vm_t - 1), Vm recurrence over t (sequential),
// elementwise over (b,c). Pure streaming, bandwidth-bound.
#define T_STEPS 500
#define N_CH    1024
#define BLOCK   256
#define TCHUNK  20                       // timesteps staged per LDS buffer
#define NCHUNK  (T_STEPS / TCHUNK)       // 25
#define THRESH  1.0f

#if __has_builtin(__builtin_amdgcn_global_load_async_to_lds_b32)
#define LIF_ASYNC 1
typedef __attribute__((address_space(1))) int as1_int;  // "__device__ int"
typedef __attribute__((address_space(3))) int as3_int;  // "__shared__ int"
#else
#define LIF_ASYNC 0
#endif

#if __has_builtin(__builtin_amdgcn_s_wait_asynccnt)
#define WAIT_ASYNC(n) __builtin_amdgcn_s_wait_asynccnt(n)
#else
#define WAIT_ASYNC(n) asm volatile("s_wait_asynccnt %0" ::"i"(n) : "memory")
#endif

__device__ __forceinline__ float lif_step(float& Vm, float xv, float wi,
                                          float decay) {
  // forget = step(1 - Vm)  ->  keep Vm only if Vm < 1 (strict)
  float keep = (Vm < THRESH) ? Vm : 0.0f;
  float v = fmaf(decay, keep, wi * xv);
  Vm = fmaxf(v, 0.0f);                      // relu
  return (Vm > THRESH) ? 1.0f : 0.0f;       // spike = step(Vm - 1)
}

__global__ __launch_bounds__(BLOCK) void lif_scan_kernel(
    const float* __restrict__ x, const float* __restrict__ w_input,
    const float* __restrict__ w_leak, float* __restrict__ out) {
  const int tid = threadIdx.x;
  const int c = blockIdx.x * BLOCK + tid;   // channel
  const int b = blockIdx.y;                 // batch
  const float wi = w_input[0];
  const float decay = 1.0f - w_leak[0];

  // element index of (b, t=0, c); totals < 2^31 so 32-bit math is safe
  const unsigned base = (unsigned)b * (unsigned)(T_STEPS * N_CH) + (unsigned)c;
  const float* __restrict__ xg = x + base;  // stride N_CH between timesteps
  float* __restrict__ yg = out + base;

  float Vm = 0.0f;

#if LIF_ASYNC
  // Double-buffered async global->LDS staging, wave-local (each lane reads
  // back only the LDS slots it loaded itself, so ASYNCcnt waits are the only
  // synchronization needed; no barriers).
  __shared__ float smem[2 * TCHUNK * BLOCK];

  // Prologue: issue chunk 0 into buffer 0 (TCHUNK async loads in flight).
#pragma unroll
  for (int u = 0; u < TCHUNK; ++u) {
    __builtin_amdgcn_global_load_async_to_lds_b32(
        (as1_int*)(xg + (unsigned)u * N_CH),
        (as3_int*)&smem[u * BLOCK + tid], 0, 0);
  }

  for (int k = 0; k < NCHUNK; ++k) {
    const int ph = k & 1;
    if (k + 1 < NCHUNK) {
      // Issue next chunk into the other buffer before consuming this one.
      const unsigned t0 = (unsigned)(k + 1) * TCHUNK;
      const int obuf = (ph ^ 1) * (TCHUNK * BLOCK);
#pragma unroll
      for (int u = 0; u < TCHUNK; ++u) {
        __builtin_amdgcn_global_load_async_to_lds_b32(
            (as1_int*)(xg + (t0 + (unsigned)u) * N_CH),
            (as3_int*)&smem[obuf + u * BLOCK + tid], 0, 0);
      }
      // Loads complete in order: <=TCHUNK outstanding means chunk k is done
      // while chunk k+1 stays in flight (double-buffer overlap).
      WAIT_ASYNC(TCHUNK);
    } else {
      WAIT_ASYNC(0);
    }

    const int ibuf = ph * (TCHUNK * BLOCK);
    const unsigned tbase = (unsigned)k * TCHUNK;
#pragma unroll
    for (int u = 0; u < TCHUNK; ++u) {
      float xv = smem[ibuf + u * BLOCK + tid];   // conflict-free: 32 consec banks
      float spike = lif_step(Vm, xv, wi, decay);
      __builtin_nontemporal_store(spike, yg + (tbase + (unsigned)u) * N_CH);
    }
  }
#else
  // Fallback: register-pipelined direct loads (TCHUNK loads batched in flight).
  float xv[TCHUNK];
  for (int k = 0; k < NCHUNK; ++k) {
    const unsigned tbase = (unsigned)k * TCHUNK;
#pragma unroll
    for (int u = 0; u < TCHUNK; ++u)
      xv[u] = __builtin_nontemporal_load(xg + (tbase + (unsigned)u) * N_CH);
#pragma unroll
    for (int u = 0; u < TCHUNK; ++u) {
      float spike = lif_step(Vm, xv[u], wi, decay);
      __builtin_nontemporal_store(spike, yg + (tbase + (unsigned)u) * N_CH);
    }
  }
#endif
}

extern "C" void kernel_launch(void* const* d_in, const int* in_sizes, int n_in,
                              void* d_out, int out_size, void* d_ws,
                              size_t ws_size, hipStream_t stream) {
  const float* x = (const float*)d_in[0];
  const float* w_input = (const float*)d_in[1];
  const float* w_leak = (const float*)d_in[2];
  float* out = (float*)d_out;

  const int Bn = in_sizes[0] / (T_STEPS * N_CH);  // 64
  dim3 grid(N_CH / BLOCK, Bn);                    // (4, 64) = 256 blocks
  lif_scan_kernel<<<grid, dim3(BLOCK), 0, stream>>>(x, w_input, w_leak, out);
}